// SparseISTAExtractor_58188216926848
// MI455X (gfx1250) — compile-verified
//
#include <hip/hip_runtime.h>
#include <hip/hip_bf16.h>

// ---------------------------------------------------------------------------
// Sparse ISTA PPR extractor for MI455X (gfx1250, wave32)
//   - sparsity-aware scatter SpMV (skip edges with q[src]==0; src stream only)
//   - exact top-512 selection via 3-level radix histogram on float bits
//   - final 512x512^T @ q matvec done with V_WMMA_F32_16X16X4_F32
//     (branch-free B fragment: no exec-mask churn inside the WMMA loop)
// ---------------------------------------------------------------------------

#define KSEL   512
#define NSTEPS 10
#define ALPHA_ 0.15f
#define THR_   (0.15f * 1.0e-3f)   // ALPHA * RHO

typedef __attribute__((ext_vector_type(2))) float v2f;
typedef __attribute__((ext_vector_type(8))) float v8f;

// ---------------- basic kernels ----------------

__global__ void k_init_q(float* __restrict__ q, const int* __restrict__ start, int n) {
    int i = blockIdx.x * blockDim.x + threadIdx.x;
    if (i < n) q[i] = (i == *start) ? 1.0f : 0.0f;
}

__global__ void k_zero(float* __restrict__ p, int n) {
    int i = blockIdx.x * blockDim.x + threadIdx.x;
    if (i < n) p[i] = 0.0f;
}

// scatter SpMV: acc[dst] += w * q[src]; skip (wave-coherently) when q[src]==0
__global__ void k_spmv(const float* __restrict__ w, const int* __restrict__ src,
                       const int* __restrict__ dst, const float* __restrict__ q,
                       float* __restrict__ acc, int e) {
    int i = blockIdx.x * blockDim.x + threadIdx.x;
    if (i >= e) return;
    int s = src[i];
    float qs = q[s];
    if (qs != 0.0f) {
        atomicAdd(&acc[dst[i]], w[i] * qs);
    }
}

// y = (1-ALPHA)*aTq - ALPHA*s   (in place; s is one-hot at start)
__global__ void k_finalize_y(float* __restrict__ y, const int* __restrict__ start, int n) {
    int i = blockIdx.x * blockDim.x + threadIdx.x;
    if (i >= n) return;
    float t = (1.0f - ALPHA_) * y[i];
    if (i == *start) t -= ALPHA_;
    y[i] = t;
}

// ---------------- exact top-K selection (3-level radix on |y| bits) --------
// state layout (unsigned words):
// [0]=prefix [1]=mask [2]=shift [3]=krem [4]=unused [5]=thresh [6]=need_eq
// [7]=nsel   [8]=neq

__global__ __launch_bounds__(1024) void k_topk_init(unsigned* __restrict__ hist,
                                                    unsigned* __restrict__ st) {
    int t = blockIdx.x * blockDim.x + threadIdx.x;
    if (t < 2048) hist[t] = 0u;
    if (t == 0) {
        st[0] = 0u; st[1] = 0u; st[2] = 21u; st[3] = (unsigned)KSEL;
        st[4] = 0u; st[5] = 0u; st[6] = 0u;  st[7] = 0u; st[8] = 0u;
    }
}

__global__ __launch_bounds__(1024) void k_histo(const float* __restrict__ srcv, int n,
                                                const unsigned* __restrict__ st,
                                                unsigned* __restrict__ hist) {
    __shared__ unsigned lh[2048];
    for (int b = threadIdx.x; b < 2048; b += blockDim.x) lh[b] = 0u;
    __syncthreads();
    unsigned prefix = st[0], mask = st[1], shift = st[2];
    int stride = gridDim.x * blockDim.x;
    for (int i = blockIdx.x * blockDim.x + threadIdx.x; i < n; i += stride) {
        unsigned u = __float_as_uint(fabsf(srcv[i]));
        if ((u & mask) == prefix) atomicAdd(&lh[(u >> shift) & 2047u], 1u);
    }
    __syncthreads();
    for (int b = threadIdx.x; b < 2048; b += blockDim.x) {
        unsigned c = lh[b];
        if (c) atomicAdd(&hist[b], c);
    }
}

__global__ void k_scan(unsigned* __restrict__ hist, unsigned* __restrict__ st) {
    if (threadIdx.x != 0 || blockIdx.x != 0) return;
    unsigned shift = st[2];
    unsigned krem  = st[3];
    unsigned cum = 0u; int bf = 0;
    for (int b = 2047; b >= 0; --b) {
        unsigned c = hist[b];
        if (cum + c >= krem) { bf = b; break; }
        cum += c;
    }
    st[3] = krem - cum;                   // still needed within chosen bin
    if (shift == 21u) {
        st[0] |= (unsigned)bf << 21; st[1] |= 0x7FFu << 21; st[2] = 10u;
    } else if (shift == 10u) {
        st[0] |= (unsigned)bf << 10; st[1] |= 0x7FFu << 10; st[2] = 0u;
    } else {
        st[5] = st[0] | (unsigned)bf;     // exact K-th largest bit pattern
        st[6] = krem - cum;               // how many ties to take
    }
    for (int b = 0; b < 2048; ++b) hist[b] = 0u;  // ready for next level
}

__global__ void k_select(const float* __restrict__ srcv, int n, unsigned* __restrict__ st,
                         int* __restrict__ idx, float* __restrict__ val) {
    int i = blockIdx.x * blockDim.x + threadIdx.x;
    if (i >= n) return;
    float v = srcv[i];
    unsigned u = __float_as_uint(fabsf(v));
    unsigned T = st[5];
    if (u > T) {
        unsigned p = atomicAdd(&st[7], 1u);
        if (p < (unsigned)KSEL) { idx[p] = i; val[p] = v; }
    } else if (u == T) {
        unsigned e = atomicAdd(&st[8], 1u);
        if (e < st[6]) {
            unsigned p = atomicAdd(&st[7], 1u);
            if (p < (unsigned)KSEL) { idx[p] = i; val[p] = v; }
        }
    }
}

// q[idx[j]] = softthresh(y[idx[j]], ALPHA*RHO)   (q pre-zeroed)
__global__ void k_scatter_q(float* __restrict__ q, const int* __restrict__ idx,
                            const float* __restrict__ val) {
    int j = blockIdx.x * blockDim.x + threadIdx.x;
    if (j >= KSEL) return;
    float v = val[j];
    float t = fabsf(v) - THR_;
    q[idx[j]] = (t > 0.0f) ? copysignf(t, v) : 0.0f;
}

// ---------------- final ordering / submatrix extraction --------------------

// bitonic sort 512 elements: desc by |val|, ties -> ascending index
__global__ __launch_bounds__(512) void k_sort512(const int* __restrict__ idx,
                                                 const float* __restrict__ val,
                                                 int* __restrict__ idxs,
                                                 float* __restrict__ vals) {
    __shared__ float ka[KSEL];
    __shared__ float sv[KSEL];
    __shared__ int   ia[KSEL];
    int t = threadIdx.x;
    float v = val[t];
    ka[t] = fabsf(v); sv[t] = v; ia[t] = idx[t];
    __syncthreads();
    for (int k = 2; k <= KSEL; k <<= 1) {
        for (int j = k >> 1; j > 0; j >>= 1) {
            int x = t ^ j;
            if (x > t) {
                bool dir = ((t & k) == 0);
                float a = ka[t], b = ka[x];
                int   iA = ia[t], iB = ia[x];
                bool before = (a > b) || (a == b && iA < iB);
                if (before != dir) {
                    ka[t] = b; ka[x] = a;
                    ia[t] = iB; ia[x] = iA;
                    float s0 = sv[t]; sv[t] = sv[x]; sv[x] = s0;
                }
            }
            __syncthreads();
        }
    }
    idxs[t] = ia[t];
    vals[t] = sv[t];
}

__global__ void k_init_pos(int* __restrict__ pos, int n) {
    int i = blockIdx.x * blockDim.x + threadIdx.x;
    if (i < n) pos[i] = -1;
}

__global__ void k_set_pos(int* __restrict__ pos, const int* __restrict__ idxs) {
    int j = blockIdx.x * blockDim.x + threadIdx.x;
    if (j < KSEL) pos[idxs[j]] = j;
}

__global__ void k_submat(const float* __restrict__ w, const int* __restrict__ src,
                         const int* __restrict__ dst, const int* __restrict__ pos,
                         float* __restrict__ sub, int e) {
    int i = blockIdx.x * blockDim.x + threadIdx.x;
    if (i >= e) return;
    int r = pos[src[i]];
    if (r < 0) return;
    int c = pos[dst[i]];
    if (c < 0) return;
    atomicAdd(&sub[r * KSEL + c], w[i]);
}

__global__ void k_gather_nf(const float* __restrict__ nf, const int* __restrict__ idxs,
                            float* __restrict__ out_nf) {
    int t = blockIdx.x * blockDim.x + threadIdx.x;   // K*64 threads
    if (t >= KSEL * 64) return;
    int j = t >> 6, c = t & 63;
    out_nf[t] = nf[(size_t)idxs[j] * 64 + c];
}

// ---------------- dense step: y = (1-a)*sub^T @ q - a*s via f32 WMMA -------
// One workgroup, 32 waves; wave w owns output rows [16w,16w+16).
// GEMM view: D(16x16) += A(16x4) * B(4x16), A = sub^T stripe, B col0 = dense_q.
// B fragment built branch-free (all lanes load dq, non-col-0 lanes masked to 0)
// so EXEC stays all-ones through the whole WMMA loop.

__global__ __launch_bounds__(1024) void k_dense_wmma(const float* __restrict__ sub,
                                                     const float* __restrict__ dq,
                                                     const int* __restrict__ idxs,
                                                     const int* __restrict__ start,
                                                     float* __restrict__ out_q,
                                                     float* __restrict__ err) {
    const int lane  = threadIdx.x & 31;
    const int wave  = threadIdx.x >> 5;        // 0..31
    const int mloc  = lane & 15;               // M within tile
    const int khalf = (lane >> 4) * 2;         // K sub-offset (0 or 2)
    const int mg    = wave * 16 + mloc;        // global output row
    const float nmask = (mloc == 0) ? 1.0f : 0.0f;   // select, not branch

    v8f c = {};
    for (int k = 0; k < KSEL; k += 4) {
        v2f a, b;
        // A = sub^T tile: A[mloc][kk] = sub[kk*512 + mg]
        a.x = sub[(size_t)(k + khalf)     * KSEL + mg];
        a.y = sub[(size_t)(k + khalf + 1) * KSEL + mg];
        // B: only column N==0 nonzero, holds dense_q; unconditional load + mask
        b.x = dq[k + khalf]     * nmask;
        b.y = dq[k + khalf + 1] * nmask;
        c = __builtin_amdgcn_wmma_f32_16x16x4_f32(
                /*neg_a=*/false, a, /*neg_b=*/false, b,
                /*c_mod=*/(short)0, c, /*reuse_a=*/false, /*reuse_b=*/false);
    }
    // C/D layout: VGPR r of lane L holds D[M = r + 8*(L/16)][N = L%16]
    if (mloc == 0) {
        int s0 = *start;
        for (int r = 0; r < 8; ++r) {
            int m = wave * 16 + r + 8 * (lane >> 4);
            float yv = (1.0f - ALPHA_) * c[r]
                     - ALPHA_ * ((idxs[m] == s0) ? 1.0f : 0.0f);
            float t  = fabsf(yv) - THR_;
            float qs = (t > 0.0f) ? copysignf(t, yv) : 0.0f;
            out_q[m] = qs;
            float d = dq[m] - qs;
            err[m] = d * d;
        }
    }
}

__global__ __launch_bounds__(512) void k_reduce512(const float* __restrict__ e,
                                                   float* __restrict__ out) {
    __shared__ float s[KSEL];
    int t = threadIdx.x;
    s[t] = e[t];
    __syncthreads();
    for (int w = 256; w > 0; w >>= 1) {
        if (t < w) s[t] += s[t + w];
        __syncthreads();
    }
    if (t == 0) *out = s[0];
}

// ---------------- host orchestration ---------------------------------------

static inline size_t align256(size_t x) { return (x + 255) & ~(size_t)255; }

extern "C" void kernel_launch(void* const* d_in, const int* in_sizes, int n_in,
                              void* d_out, int out_size, void* d_ws, size_t ws_size,
                              hipStream_t stream) {
    const float* edge_data  = (const float*)d_in[0];
    const float* node_feats = (const float*)d_in[1];
    const int*   edge_src   = (const int*)d_in[2];
    const int*   edge_dst   = (const int*)d_in[3];
    const int*   start      = (const int*)d_in[4];

    const int E = in_sizes[0];
    const int N = in_sizes[1] / 64;

    // workspace carve-up (~6 MB)
    char* ws = (char*)d_ws;
    size_t o = 0;
    float*    q    = (float*)(ws + o);    o = align256(o + 4 * (size_t)N);
    float*    y    = (float*)(ws + o);    o = align256(o + 4 * (size_t)N);
    int*      pos  = (int*)(ws + o);      o = align256(o + 4 * (size_t)N);
    unsigned* hist = (unsigned*)(ws + o); o = align256(o + 4 * 2048);
    unsigned* st   = (unsigned*)(ws + o); o = align256(o + 64);
    int*      idx  = (int*)(ws + o);      o = align256(o + 4 * KSEL);
    float*    val  = (float*)(ws + o);    o = align256(o + 4 * KSEL);
    int*      idxs = (int*)(ws + o);      o = align256(o + 4 * KSEL);
    float*    vals = (float*)(ws + o);    o = align256(o + 4 * KSEL);
    float*    e512 = (float*)(ws + o);    o = align256(o + 4 * KSEL);

    float* out      = (float*)d_out;
    float* out_q    = out;                         // 512
    float* out_nf   = out + KSEL;                  // 512*64
    float* out_sub  = out + KSEL + KSEL * 64;      // 512*512
    float* out_err  = out_sub + KSEL * KSEL;       // 1 (contiguous after submat)

    const int nbN = (N + 255) / 256;
    const int nbE = (E + 255) / 256;

    k_init_q<<<nbN, 256, 0, stream>>>(q, start, N);

    for (int it = 0; it < NSTEPS; ++it) {
        k_zero<<<nbN, 256, 0, stream>>>(y, N);
        k_spmv<<<nbE, 256, 0, stream>>>(edge_data, edge_src, edge_dst, q, y, E);
        k_finalize_y<<<nbN, 256, 0, stream>>>(y, start, N);

        k_topk_init<<<2, 1024, 0, stream>>>(hist, st);
        for (int lvl = 0; lvl < 3; ++lvl) {
            k_histo<<<128, 1024, 0, stream>>>(y, N, st, hist);
            k_scan<<<1, 1, 0, stream>>>(hist, st);
        }
        k_select<<<nbN, 256, 0, stream>>>(y, N, st, idx, val);

        k_zero<<<nbN, 256, 0, stream>>>(q, N);
        k_scatter_q<<<2, 256, 0, stream>>>(q, idx, val);
    }

    // final support: top-K of |q|
    k_topk_init<<<2, 1024, 0, stream>>>(hist, st);
    for (int lvl = 0; lvl < 3; ++lvl) {
        k_histo<<<128, 1024, 0, stream>>>(q, N, st, hist);
        k_scan<<<1, 1, 0, stream>>>(hist, st);
    }
    k_select<<<nbN, 256, 0, stream>>>(q, N, st, idx, val);
    k_sort512<<<1, 512, 0, stream>>>(idx, val, idxs, vals);

    // K x K submatrix extraction
    k_init_pos<<<nbN, 256, 0, stream>>>(pos, N);
    k_set_pos<<<2, 256, 0, stream>>>(pos, idxs);
    k_zero<<<(KSEL * KSEL + 1 + 255) / 256, 256, 0, stream>>>(out_sub, KSEL * KSEL + 1);
    k_submat<<<nbE, 256, 0, stream>>>(edge_data, edge_src, edge_dst, pos, out_sub, E);

    // node features
    k_gather_nf<<<(KSEL * 64 + 255) / 256, 256, 0, stream>>>(node_feats, idxs, out_nf);

    // dense implicit-layer step via f32 WMMA + deterministic error reduction
    k_dense_wmma<<<1, 1024, 0, stream>>>(out_sub, vals, idxs, start, out_q, e512);
    k_reduce512<<<1, 512, 0, stream>>>(e512, out_err);
}